// SAGE_5789615915310
// MI455X (gfx1250) — compile-verified
//
#include <hip/hip_runtime.h>
#include <hip/hip_bf16.h>

typedef __attribute__((ext_vector_type(2))) float v2f;
typedef __attribute__((ext_vector_type(8))) float v8f;

#define D 128  // feature dim (both layers)

// ---------------- zero fill (float4 granular) ----------------
__global__ void zero_f4_kernel(float4* __restrict__ p, long n4) {
    long i = (long)blockIdx.x * blockDim.x + threadIdx.x;
    if (i < n4) p[i] = make_float4(0.f, 0.f, 0.f, 0.f);
}

// ---------------- in-degree count ----------------
__global__ void deg_kernel(const int* __restrict__ edst, float* __restrict__ deg, int E) {
    int e = blockIdx.x * blockDim.x + threadIdx.x;
    if (e < E) atomicAdd(&deg[edst[e]], 1.0f);
}

// ---------------- deg -> 1/max(deg,1) in place ----------------
__global__ void invdeg_kernel(float* __restrict__ deg, int N) {
    int i = blockIdx.x * blockDim.x + threadIdx.x;
    if (i < N) deg[i] = 1.0f / fmaxf(deg[i], 1.0f);
}

// ---------------- edge gather + scatter-add (segment_sum) ----------------
// one wave32 per edge; lane handles 4 consecutive floats of the 128-float row
__global__ void scatter_kernel(const float* __restrict__ feat,
                               const int* __restrict__ esrc,
                               const int* __restrict__ edst,
                               float* __restrict__ agg, int E) {
    int t = blockIdx.x * blockDim.x + threadIdx.x;
    int e = t >> 5;
    int lane = t & 31;
    if (e >= E) return;
    int s = esrc[e];
    int d = edst[e];
    const float4 v = *(const float4*)(feat + (size_t)s * D + lane * 4);
    float* dst = agg + (size_t)d * D + lane * 4;
    atomicAdd(dst + 0, v.x);
    atomicAdd(dst + 1, v.y);
    atomicAdd(dst + 2, v.z);
    atomicAdd(dst + 3, v.w);
}

// ---------------- fused SAGE layer GEMM ----------------
// Out[m,n] = act( sum_k A_self[m,k]*W_self[n,k]
//               + inv_deg[m]*sum_k A_agg[m,k]*W_neigh[n,k] + bias[n] )
// block = 256 threads = 8 waves; wave w owns cols [16w,16w+16); block owns 16 rows.
// V_WMMA_F32_16X16X4_F32, K stepped by 4; dual accumulators so the inner loop
// is pure load + wmma (inv_deg scale distributed to the epilogue).
__global__ __launch_bounds__(256) void sage_gemm_kernel(
        const float* __restrict__ A_self, const float* __restrict__ A_agg,
        const float* __restrict__ inv_deg,
        const float* __restrict__ W_self, const float* __restrict__ W_neigh,
        const float* __restrict__ bias,
        float* __restrict__ Out, int N, int do_relu) {
    const int lane = threadIdx.x & 31;
    const int wave = threadIdx.x >> 5;
    const int m = lane & 15;        // row-in-tile for A, col-in-tile for B/C
    const int khalf = lane >> 4;    // 0: K pair {0,1}; 1: K pair {2,3}
    const int rowBase = blockIdx.x * 16;
    const int colBase = wave * 16;

    // clamp A-row for safety (EXEC must stay all-1s through the WMMA loop)
    int rowA = rowBase + m;
    if (rowA >= N) rowA = N - 1;
    const float scale_m = inv_deg[rowA];   // lane m holds scale for row rowBase+m

    const float* pAs = A_self + (size_t)rowA * D;
    const float* pAn = A_agg + (size_t)rowA * D;
    const float* pWs = W_self + (size_t)(colBase + m) * D;   // W[n][k], row-major
    const float* pWn = W_neigh + (size_t)(colBase + m) * D;

    v8f cs = {};
    v8f cn = {};
#pragma unroll
    for (int k0 = 0; k0 < D; k0 += 4) {
        const int kk = k0 + 2 * khalf;
        v2f a_s = *(const v2f*)(pAs + kk);
        v2f a_n = *(const v2f*)(pAn + kk);
        v2f b_s = *(const v2f*)(pWs + kk);
        v2f b_n = *(const v2f*)(pWn + kk);
        // D = A*B + C  (8 args: neg_a, A, neg_b, B, c_mod, C, reuse_a, reuse_b)
        cs = __builtin_amdgcn_wmma_f32_16x16x4_f32(false, a_s, false, b_s,
                                                   (short)0, cs, false, false);
        cn = __builtin_amdgcn_wmma_f32_16x16x4_f32(false, a_n, false, b_n,
                                                   (short)0, cn, false, false);
    }

    const float bv = bias[colBase + m];
#pragma unroll
    for (int v = 0; v < 8; ++v) {
        int rowOut = rowBase + v + 8 * khalf;   // C layout: lanes>=16 hold M=v+8
        // cross-lane fetch of inv_deg for this output row (lane r holds row rowBase+r)
        float sc = __shfl(scale_m, v + 8 * khalf, 32);
        if (rowOut < N) {
            float val = cs[v] + sc * cn[v] + bv;
            if (do_relu) val = fmaxf(val, 0.0f);
            Out[(size_t)rowOut * D + colBase + m] = val;
        }
    }
}

extern "C" void kernel_launch(void* const* d_in, const int* in_sizes, int n_in,
                              void* d_out, int out_size, void* d_ws, size_t ws_size,
                              hipStream_t stream) {
    const float* x   = (const float*)d_in[0];
    const float* Ws0 = (const float*)d_in[1];
    const float* Wn0 = (const float*)d_in[2];
    const float* b0  = (const float*)d_in[3];
    const float* Ws1 = (const float*)d_in[4];
    const float* Wn1 = (const float*)d_in[5];
    const float* b1  = (const float*)d_in[6];
    const int* esrc  = (const int*)d_in[7];
    const int* edst  = (const int*)d_in[8];
    float* out = (float*)d_out;

    const int N = in_sizes[0] / D;   // 50000
    const int E = in_sizes[7];       // 800000

    // workspace: [deg|inv_deg (padded)] [agg N*D] [h N*D]
    const size_t degPad = ((size_t)N + 127) & ~(size_t)127;
    float* deg = (float*)d_ws;
    float* agg = deg + degPad;
    float* h   = agg + (size_t)N * D;

    const int T = 256;
    const long zero0 = (long)(degPad + (size_t)N * D) / 4;  // deg+agg, contiguous
    const long zeroA = (long)((size_t)N * D) / 4;           // agg only
    const int rowTiles = (N + 15) / 16;

    // ---- degree + inv_deg ----
    zero_f4_kernel<<<(int)((zero0 + T - 1) / T), T, 0, stream>>>((float4*)deg, zero0);
    deg_kernel<<<(E + T - 1) / T, T, 0, stream>>>(edst, deg, E);
    invdeg_kernel<<<(N + T - 1) / T, T, 0, stream>>>(deg, N);

    // ---- layer 0 ----
    scatter_kernel<<<(E * 32 + T - 1) / T, T, 0, stream>>>(x, esrc, edst, agg, E);
    sage_gemm_kernel<<<rowTiles, T, 0, stream>>>(x, agg, deg, Ws0, Wn0, b0, h, N, 1);

    // ---- layer 1 (reuse agg buffer) ----
    zero_f4_kernel<<<(int)((zeroA + T - 1) / T), T, 0, stream>>>((float4*)agg, zeroA);
    scatter_kernel<<<(E * 32 + T - 1) / T, T, 0, stream>>>(h, esrc, edst, agg, E);
    sage_gemm_kernel<<<rowTiles, T, 0, stream>>>(h, agg, deg, Ws1, Wn1, b1, out, N, 0);
}